// DecoupledMlaRopeAttention_6193342841649
// MI455X (gfx1250) — compile-verified
//
#include <hip/hip_runtime.h>
#include <math.h>

// ---------------- problem constants ----------------
constexpr int DIMC   = 2048;
constexpr int SLEN   = 2048;
constexpr int BATCH  = 2;
constexpr int NH     = 16;   // heads
constexpr int NKV    = 4;    // kv heads
constexpr int CQD    = 128;
constexpr int QND    = 96;
constexpr int QRD    = 32;
constexpr int CKVD   = 128;
constexpr int KND    = 64;
constexpr int KRD    = 64;
constexpr int VDD    = 256;
constexpr int DQK    = 128;  // QN+QR == KN+KR
constexpr int ROWS   = BATCH * SLEN;             // 4096
constexpr int UQ_N   = NH * (QND + QRD);         // 2048
constexpr int DKV_N  = CKVD + KRD;               // 192
constexpr int UKV_N  = NKV * (KND + VDD);        // 1280
constexpr int HVD    = NH * VDD;                 // 4096

typedef float v2f __attribute__((ext_vector_type(2)));
typedef float v8f __attribute__((ext_vector_type(8)));

__device__ __forceinline__ v8f wmma_f32(v2f a, v2f b, v8f c) {
  // D = A(16x4 f32) * B(4x16 f32) + C(16x16 f32)  -> v_wmma_f32_16x16x4_f32
  return __builtin_amdgcn_wmma_f32_16x16x4_f32(
      /*neg_a=*/false, a, /*neg_b=*/false, b,
      /*c_mod=*/(short)0, c, /*reuse_a=*/false, /*reuse_b=*/false);
}

__device__ __forceinline__ v8f v8f_zero() {
  v8f z = {0.f, 0.f, 0.f, 0.f, 0.f, 0.f, 0.f, 0.f};
  return z;
}

// ---------------------------------------------------------------------------
// LDS-staged, double-buffered fp32 WMMA GEMM: C[M,N] = A[M,K] @ B[K,N].
// Block = 256 threads (8 waves) -> 128(M) x 64(N) tile; K chunk = 32.
// B is stored K-pair-interleaved so each fragment is a single ds_load_b64.
// Requires M%128==0, N%64==0, K%32==0 (true for every GEMM here).
// ---------------------------------------------------------------------------
constexpr int KC = 32;

__global__ __launch_bounds__(256) void gemm_wmma_f32_lds(
    const float* __restrict__ A, const float* __restrict__ B,
    float* __restrict__ C, int M, int N, int Kd) {
  __shared__ float As[2][128][36];          // stride 36 -> 144B rows (16B aligned)
  __shared__ float Bs[2][KC / 2][64][2];    // [kpair][n][k&1]

  const int tid  = threadIdx.x;
  const int lane = tid & 31;
  const int wave = tid >> 5;
  const int r    = lane & 15;
  const int hf   = lane >> 4;
  const int m0b  = blockIdx.x * 128;
  const int n0   = blockIdx.y * 64;

  v8f acc[4];
#pragma unroll
  for (int j = 0; j < 4; ++j) acc[j] = v8f_zero();

  const int aRow[4] = {0 * 32 + (tid >> 3), 1 * 32 + (tid >> 3),
                       2 * 32 + (tid >> 3), 3 * 32 + (tid >> 3)};
  const int aCol = (tid * 4) & 31;
  const int bKr[2] = {0 * 16 + (tid >> 4), 1 * 16 + (tid >> 4)};
  const int bNc = (tid * 4) & 63;

  const int nchunk = Kd / KC;
  float4 aReg[4], bReg[2];

  // ---- prologue: load chunk 0 straight into buffer 0 ----
#pragma unroll
  for (int rep = 0; rep < 4; ++rep)
    aReg[rep] = *(const float4*)(A + (size_t)(m0b + aRow[rep]) * Kd + aCol);
#pragma unroll
  for (int rep = 0; rep < 2; ++rep)
    bReg[rep] = *(const float4*)(B + (size_t)bKr[rep] * N + n0 + bNc);
#pragma unroll
  for (int rep = 0; rep < 4; ++rep)
    *(float4*)&As[0][aRow[rep]][aCol] = aReg[rep];
#pragma unroll
  for (int rep = 0; rep < 2; ++rep) {
    const int kp = bKr[rep] >> 1, kb = bKr[rep] & 1;
    Bs[0][kp][bNc + 0][kb] = bReg[rep].x;
    Bs[0][kp][bNc + 1][kb] = bReg[rep].y;
    Bs[0][kp][bNc + 2][kb] = bReg[rep].z;
    Bs[0][kp][bNc + 3][kb] = bReg[rep].w;
  }
  __syncthreads();

  for (int c = 0; c < nchunk; ++c) {
    const int buf = c & 1;
    // ---- prefetch chunk c+1 into registers (overlaps WMMA stream) ----
    if (c + 1 < nchunk) {
      const int kc = (c + 1) * KC;
#pragma unroll
      for (int rep = 0; rep < 4; ++rep)
        aReg[rep] =
            *(const float4*)(A + (size_t)(m0b + aRow[rep]) * Kd + kc + aCol);
#pragma unroll
      for (int rep = 0; rep < 2; ++rep)
        bReg[rep] =
            *(const float4*)(B + (size_t)(kc + bKr[rep]) * N + n0 + bNc);
    }
    // ---- compute this chunk from LDS: 8 K-steps x 4 N-tiles = 32 WMMAs ----
#pragma unroll
    for (int k = 0; k < KC; k += 4) {
      v2f a = *(const v2f*)&As[buf][wave * 16 + r][k + 2 * hf];
#pragma unroll
      for (int j = 0; j < 4; ++j) {
        v2f bb = *(const v2f*)&Bs[buf][(k + 2 * hf) >> 1][j * 16 + r][0];
        acc[j] = wmma_f32(a, bb, acc[j]);
      }
    }
    // ---- stage prefetched chunk into the other buffer ----
    if (c + 1 < nchunk) {
      const int nbuf = buf ^ 1;
#pragma unroll
      for (int rep = 0; rep < 4; ++rep)
        *(float4*)&As[nbuf][aRow[rep]][aCol] = aReg[rep];
#pragma unroll
      for (int rep = 0; rep < 2; ++rep) {
        const int kp = bKr[rep] >> 1, kb = bKr[rep] & 1;
        Bs[nbuf][kp][bNc + 0][kb] = bReg[rep].x;
        Bs[nbuf][kp][bNc + 1][kb] = bReg[rep].y;
        Bs[nbuf][kp][bNc + 2][kb] = bReg[rep].z;
        Bs[nbuf][kp][bNc + 3][kb] = bReg[rep].w;
      }
    }
    __syncthreads();
  }

#pragma unroll
  for (int j = 0; j < 4; ++j)
#pragma unroll
    for (int i = 0; i < 8; ++i)
      C[(size_t)(m0b + wave * 16 + i + 8 * hf) * N + n0 + j * 16 + r] =
          acc[j][i];
}

// ---------------------------------------------------------------------------
// T5 layernorm: out[row,0:128] = (x-mu)*rsqrt(var+1e-5)*gamma ; wave per row.
// ---------------------------------------------------------------------------
__global__ __launch_bounds__(32) void t5ln_kernel(
    const float* __restrict__ in, int in_stride,
    const float* __restrict__ gamma, float* __restrict__ out) {
  const int row  = blockIdx.x;
  const int lane = threadIdx.x;
  const float* ip = in + (size_t)row * in_stride;
  float v[4], s = 0.f, s2 = 0.f;
#pragma unroll
  for (int i = 0; i < 4; ++i) {
    v[i] = ip[lane + 32 * i];
    s  += v[i];
    s2 += v[i] * v[i];
  }
#pragma unroll
  for (int off = 1; off < 32; off <<= 1) {
    s  += __shfl_xor(s, off, 32);
    s2 += __shfl_xor(s2, off, 32);
  }
  const float mu  = s * (1.f / 128.f);
  const float var = s2 * (1.f / 128.f) - mu * mu;
  const float rs  = rsqrtf(var + 1e-5f);
  float* op = out + (size_t)row * 128;
#pragma unroll
  for (int i = 0; i < 4; ++i)
    op[lane + 32 * i] = (v[i] - mu) * rs * gamma[lane + 32 * i];
}

// ---------------------------------------------------------------------------
// RoPE helper: interleaved rope, D = rope dim, pair index, position s.
// ---------------------------------------------------------------------------
__device__ __forceinline__ void rope_pair(float spos, int pair, float invD,
                                          float xe, float xo,
                                          float& re, float& ro) {
  const float lnTheta = 9.210340371976184f;  // ln(10000)
  float freq = expf(-(2.0f * (float)pair) * invD * lnTheta);
  float ang  = spos * freq;
  float c = cosf(ang), sn = sinf(ang);
  re = xe * c - xo * sn;
  ro = xe * sn + xo * c;
}

// Build q_states (B,H,S,128): nope cols 0..95 copied, rope cols 96..127.
__global__ __launch_bounds__(256) void build_q_kernel(
    const float* __restrict__ qraw, float* __restrict__ qst) {
  size_t idx = (size_t)blockIdx.x * blockDim.x + threadIdx.x;
  const size_t total = (size_t)BATCH * NH * SLEN * DQK;
  if (idx >= total) return;
  int j = idx & 127;
  int s = (int)((idx >> 7) & (SLEN - 1));
  int h = (int)((idx >> 18) & (NH - 1));
  int b = (int)(idx >> 22);
  const float* qr = qraw + ((size_t)(b * SLEN + s)) * UQ_N;
  float val;
  if (j < QND) {
    val = qr[h * QND + j];
  } else {
    int jj = j - QND;
    int pair = jj >> 1;
    const float* base = qr + NH * QND + h * QRD;
    float xe = base[2 * pair], xo = base[2 * pair + 1];
    float re, ro;
    rope_pair((float)s, pair, 1.0f / (float)QRD, xe, xo, re, ro);
    val = (jj & 1) ? ro : re;
  }
  qst[idx] = val;
}

// Build k_states (B,KV,S,128): nope 0..63 from kvraw, rope 64..127 broadcast.
__global__ __launch_bounds__(256) void build_k_kernel(
    const float* __restrict__ kvraw, const float* __restrict__ ckv_kr,
    float* __restrict__ kst) {
  size_t idx = (size_t)blockIdx.x * blockDim.x + threadIdx.x;
  const size_t total = (size_t)BATCH * NKV * SLEN * DQK;
  if (idx >= total) return;
  int j   = idx & 127;
  int s   = (int)((idx >> 7) & (SLEN - 1));
  int kvh = (int)((idx >> 18) & (NKV - 1));
  int b   = (int)(idx >> 20);
  size_t row = (size_t)(b * SLEN + s);
  float val;
  if (j < KND) {
    val = kvraw[row * UKV_N + kvh * KND + j];
  } else {
    int jj = j - KND;
    int pair = jj >> 1;
    const float* base = ckv_kr + row * DKV_N + CKVD;
    float xe = base[2 * pair], xo = base[2 * pair + 1];
    float re, ro;
    rope_pair((float)s, pair, 1.0f / (float)KRD, xe, xo, re, ro);
    val = (jj & 1) ? ro : re;
  }
  kst[idx] = val;
}

// Build v_states in K-PAIR-INTERLEAVED layout:
//   VPI[(b*NKV+kvh)][s>>1][d][s&1]  (per head: (SLEN/2) x VDD x 2)
// so a PV WMMA B-fragment (V[s][n], V[s+1][n]) is one aligned b64 load that
// is also contiguous across the 16 fragment lanes.
__global__ __launch_bounds__(256) void build_v_kernel(
    const float* __restrict__ kvraw, float* __restrict__ vpi) {
  size_t idx = (size_t)blockIdx.x * blockDim.x + threadIdx.x;
  const size_t total = (size_t)BATCH * NKV * SLEN * VDD;
  if (idx >= total) return;
  int d   = idx & 255;
  int s   = (int)((idx >> 8) & (SLEN - 1));
  int kvh = (int)((idx >> 19) & (NKV - 1));
  int b   = (int)(idx >> 21);
  size_t row = (size_t)(b * SLEN + s);
  float val = kvraw[row * UKV_N + NKV * KND + kvh * VDD + d];
  size_t dst = ((size_t)(b * NKV + kvh) * (SLEN / 2) + (s >> 1)) * (2 * VDD) +
               d * 2 + (s & 1);
  vpi[dst] = val;
}

// ---------------------------------------------------------------------------
// Causal flash attention. grid = (B*H, S/128), block = 256 (8 waves).
// Each wave owns 16 query rows and streams K/V tiles of 16 up to the diagonal.
// V is consumed from the pair-interleaved layout (single b64 per fragment).
// No block barriers (waves have different causal trip counts).
// ---------------------------------------------------------------------------
__global__ __launch_bounds__(256) void attn_kernel(
    const float* __restrict__ Q, const float* __restrict__ K,
    const float* __restrict__ V, float* __restrict__ O) {
  __shared__ float ps[8][16][16];

  const int lane = threadIdx.x & 31;
  const int wave = threadIdx.x >> 5;
  const int r    = lane & 15;
  const int hf   = lane >> 4;
  const int bh   = blockIdx.x;
  const int b    = bh >> 4;
  const int h    = bh & 15;
  const int kvh  = h >> 2;
  const int q0   = blockIdx.y * 128 + wave * 16;

  const float* Qb = Q + ((size_t)bh * SLEN + q0) * DQK;
  const float* Kb = K + ((size_t)(b * NKV + kvh) * SLEN) * DQK;
  const float* Vb = V + ((size_t)(b * NKV + kvh) * SLEN) * VDD;  // pair layout

  const float scale = 0.08838834764831845f;  // 1/sqrt(128)

  v2f qa[32];
#pragma unroll
  for (int di = 0; di < 32; ++di) {
    const float* p = Qb + (size_t)r * DQK + di * 4 + 2 * hf;
    v2f t;
    t.x = p[0] * scale;
    t.y = p[1] * scale;
    qa[di] = t;
  }

  float m_i[8], l_i[8];
#pragma unroll
  for (int i = 0; i < 8; ++i) { m_i[i] = -1e30f; l_i[i] = 0.f; }
  v8f acc[16];
#pragma unroll
  for (int nt = 0; nt < 16; ++nt) acc[nt] = v8f_zero();

  const int ktmax = (q0 + 15) >> 4;
  for (int kt = 0; kt <= ktmax; ++kt) {
    const int k0 = kt * 16;

    // ---- prefetch next K tile (16x128) and next V tile (16x256) into L2/L0
    if (kt < ktmax) {
      const float* kpf = Kb + (size_t)(k0 + 16 + r) * DQK + hf * 64;
      __builtin_prefetch(kpf, 0, 3);                 // global_prefetch_b8
      const float* vpf = Vb + (size_t)((k0 + 16) >> 1) * (2 * VDD) + lane * 64;
      __builtin_prefetch(vpf, 0, 3);
    }

    // ---- S = Q @ K^T (16x16) over d=128: 32 chained f32 WMMAs ----
    v8f sc = v8f_zero();
    const float* Kt = Kb + (size_t)(k0 + r) * DQK + 2 * hf;
#pragma unroll
    for (int di = 0; di < 32; ++di) {
      v2f bb;
      bb.x = Kt[di * 4];
      bb.y = Kt[di * 4 + 1];
      sc = wmma_f32(qa[di], bb, sc);
    }

    // ---- causal mask + streaming softmax ----
#pragma unroll
    for (int i = 0; i < 8; ++i) {
      const int qrow = q0 + i + 8 * hf;
      const int kcol = k0 + r;
      float v = (kcol > qrow) ? -1e30f : sc[i];
      float mx = v;
#pragma unroll
      for (int off = 1; off < 16; off <<= 1)
        mx = fmaxf(mx, __shfl_xor(mx, off, 32));
      const float mnew  = fmaxf(m_i[i], mx);
      const float alpha = __expf(m_i[i] - mnew);
      const float pe    = __expf(v - mnew);
      float rs = pe;
#pragma unroll
      for (int off = 1; off < 16; off <<= 1)
        rs += __shfl_xor(rs, off, 32);
      l_i[i] = l_i[i] * alpha + rs;
      m_i[i] = mnew;
#pragma unroll
      for (int nt = 0; nt < 16; ++nt) acc[nt][i] *= alpha;
      ps[wave][i + 8 * hf][r] = pe;
    }
    asm volatile("s_wait_dscnt 0" ::: "memory");  // wave-local LDS RAW fence

    // ---- P A-fragments ----
    v2f pa[4];
#pragma unroll
    for (int ks = 0; ks < 4; ++ks) {
      v2f t;
      t.x = ps[wave][r][ks * 4 + 2 * hf];
      t.y = ps[wave][r][ks * 4 + 1 + 2 * hf];
      pa[ks] = t;
    }

    // ---- O += P @ V_tile : 16 column tiles of VD=256, b64 V fragments ----
#pragma unroll
    for (int nt = 0; nt < 16; ++nt) {
#pragma unroll
      for (int ks = 0; ks < 4; ++ks) {
        const int kk = k0 + ks * 4 + 2 * hf;   // even
        const float* Vp =
            Vb + (size_t)(kk >> 1) * (2 * VDD) + (nt * 16 + r) * 2;
        v2f bb = *(const v2f*)Vp;
        acc[nt] = wmma_f32(pa[ks], bb, acc[nt]);
      }
    }
  }

  // ---- epilogue: divide by l, scatter to (B,S,H*VD) ----
#pragma unroll
  for (int i = 0; i < 8; ++i) {
    const float inv = 1.0f / l_i[i];
    const int qrow = q0 + i + 8 * hf;
    float* Op = O + ((size_t)(b * SLEN + qrow)) * HVD + h * VDD;
#pragma unroll
    for (int nt = 0; nt < 16; ++nt) Op[nt * 16 + r] = acc[nt][i] * inv;
  }
}

// ---------------------------------------------------------------------------
// Host-side launch
// ---------------------------------------------------------------------------
extern "C" void kernel_launch(void* const* d_in, const int* in_sizes, int n_in,
                              void* d_out, int out_size, void* d_ws,
                              size_t ws_size, hipStream_t stream) {
  (void)in_sizes; (void)n_in; (void)out_size; (void)ws_size;
  const float* x        = (const float*)d_in[0];  // (2,2048,2048)
  const float* w_dq     = (const float*)d_in[1];  // (2048,128)
  const float* gamma_q  = (const float*)d_in[2];  // (128,)
  const float* w_uq     = (const float*)d_in[3];  // (128,2048)
  const float* w_dkv    = (const float*)d_in[4];  // (2048,192)
  const float* gamma_kv = (const float*)d_in[5];  // (128,)
  const float* w_ukv    = (const float*)d_in[6];  // (128,1280)
  const float* w_o      = (const float*)d_in[7];  // (4096,2048)
  float* out = (float*)d_out;                     // (2,2048,2048)

  float* ws = (float*)d_ws;
  float* tmp_q   = ws;                               // 4096*128
  float* cq      = tmp_q   + (size_t)ROWS * CQD;     // 4096*128
  float* qraw    = cq      + (size_t)ROWS * CQD;     // 4096*2048
  float* qst     = qraw    + (size_t)ROWS * UQ_N;    // 2*16*2048*128
  float* ckv_kr  = qst     + (size_t)BATCH * NH * SLEN * DQK;   // 4096*192
  float* ckv     = ckv_kr  + (size_t)ROWS * DKV_N;   // 4096*128
  float* kvraw   = ckv     + (size_t)ROWS * CKVD;    // 4096*1280
  float* kst     = kvraw   + (size_t)ROWS * UKV_N;   // 2*4*2048*128
  float* vpi     = kst     + (size_t)BATCH * NKV * SLEN * DQK;  // 2*4*2048*256
  float* attn_o  = vpi     + (size_t)BATCH * NKV * SLEN * VDD;  // 4096*4096

  // 1) tmp_q = x @ w_dq   (4096 x 2048 x 128)
  gemm_wmma_f32_lds<<<dim3(ROWS / 128, CQD / 64), 256, 0, stream>>>(
      x, w_dq, tmp_q, ROWS, CQD, DIMC);
  // 2) cq = T5LN(tmp_q) * gamma_q
  t5ln_kernel<<<ROWS, 32, 0, stream>>>(tmp_q, CQD, gamma_q, cq);
  // 3) qraw = cq @ w_uq   (4096 x 128 x 2048)
  gemm_wmma_f32_lds<<<dim3(ROWS / 128, UQ_N / 64), 256, 0, stream>>>(
      cq, w_uq, qraw, ROWS, UQ_N, CQD);
  // 4) q_states (B,H,S,128) with RoPE on decoupled dims
  {
    size_t total = (size_t)BATCH * NH * SLEN * DQK;
    build_q_kernel<<<(unsigned)((total + 255) / 256), 256, 0, stream>>>(qraw, qst);
  }
  // 5) ckv_kr = x @ w_dkv (4096 x 2048 x 192)
  gemm_wmma_f32_lds<<<dim3(ROWS / 128, DKV_N / 64), 256, 0, stream>>>(
      x, w_dkv, ckv_kr, ROWS, DKV_N, DIMC);
  // 6) ckv = T5LN(ckv_kr[:, :128]) * gamma_kv
  t5ln_kernel<<<ROWS, 32, 0, stream>>>(ckv_kr, DKV_N, gamma_kv, ckv);
  // 7) kvraw = ckv @ w_ukv (4096 x 128 x 1280)
  gemm_wmma_f32_lds<<<dim3(ROWS / 128, UKV_N / 64), 256, 0, stream>>>(
      ckv, w_ukv, kvraw, ROWS, UKV_N, CQD);
  // 8) k_states / v_states (pair-interleaved V)
  {
    size_t tk = (size_t)BATCH * NKV * SLEN * DQK;
    build_k_kernel<<<(unsigned)((tk + 255) / 256), 256, 0, stream>>>(kvraw, ckv_kr, kst);
    size_t tv = (size_t)BATCH * NKV * SLEN * VDD;
    build_v_kernel<<<(unsigned)((tv + 255) / 256), 256, 0, stream>>>(kvraw, vpi);
  }
  // 9) causal flash attention -> attn_o (B,S,H*VD)
  attn_kernel<<<dim3(BATCH * NH, SLEN / 128), 256, 0, stream>>>(qst, kst, vpi, attn_o);
  // 10) out = attn_o @ w_o (4096 x 4096 x 2048)
  gemm_wmma_f32_lds<<<dim3(ROWS / 128, DIMC / 64), 256, 0, stream>>>(
      attn_o, w_o, out, ROWS, DIMC, HVD);
}